// Head_15032385536604
// MI455X (gfx1250) — compile-verified
//
#include <hip/hip_runtime.h>
#include <math.h>

#define Bsz 8
#define Tsz 2048
#define Csz 1024
#define Hsz 64

typedef __attribute__((ext_vector_type(2))) float v2f;
typedef __attribute__((ext_vector_type(8))) float v8f;

// Full-precision fp32 WMMA: D(16x16,f32) = A(16x4,f32) * B(4x16,f32) + C
__device__ __forceinline__ v8f wmma_f32(v2f a, v2f b, v8f c) {
    return __builtin_amdgcn_wmma_f32_16x16x4_f32(
        /*neg_a=*/false, a, /*neg_b=*/false, b,
        /*c_mod=*/(short)0, c, /*reuse_a=*/false, /*reuse_b=*/false);
}

// ---------------------------------------------------------------------------
// Stage 0: transpose the three weight matrices [C,H] -> [H,C] so the GEMM
// B-fragment loads (K-dim contiguous) become single b64 loads. 768 KB total,
// L2-resident afterwards.
// ---------------------------------------------------------------------------
__global__ __launch_bounds__(256) void wt_kernel(
    const float* __restrict__ Wq,
    const float* __restrict__ Wk,
    const float* __restrict__ Wv,
    float* __restrict__ Wt)          // [3][H][C]
{
    const int which = blockIdx.y;
    const float* W = (which == 0) ? Wq : (which == 1) ? Wk : Wv;
    float* dst = Wt + (size_t)which * Hsz * Csz;
    const int h = blockIdx.x;        // 0..63
    for (int c = threadIdx.x; c < Csz; c += 256)
        dst[(size_t)h * Csz + c] = W[(size_t)c * Hsz + h];
}

// ---------------------------------------------------------------------------
// Stage 1: Q/K/V projections.  [B*T, C] x [C, H] -> [B*T, H]  (x3 via grid.y)
// Block = 128 (4 waves); each wave computes a 16x64 tile (4 accumulators):
// per 4 WMMAs -> 1 A b64-load + 4 B b64-loads.  V is written TRANSPOSED as
// [B][H][T] so the attention P@V B-fragments are contiguous.
// ---------------------------------------------------------------------------
__global__ __launch_bounds__(128) void qkv_kernel(
    const float* __restrict__ x,
    const float* __restrict__ Wt,    // [3][H][C]
    float* __restrict__ qkv)         // Q,K: [B*T][H]; V: [B][H][T]
{
    const int lane = threadIdx.x & 31;
    const int wv   = threadIdx.x >> 5;       // wave id 0..3 -> row sub-tile
    const int half = lane >> 4;
    const int l16  = lane & 15;

    const long mrow  = (long)blockIdx.x * 64 + wv * 16;   // row tile base
    const int  which = blockIdx.y;                        // 0=Q, 1=K, 2=V
    const float* W = Wt + (size_t)which * Hsz * Csz;

    const float* arow = x + (mrow + l16) * (size_t)Csz + 2 * half;
    const float* b0 = W + (size_t)( 0 + l16) * Csz + 2 * half;
    const float* b1 = W + (size_t)(16 + l16) * Csz + 2 * half;
    const float* b2 = W + (size_t)(32 + l16) * Csz + 2 * half;
    const float* b3 = W + (size_t)(48 + l16) * Csz + 2 * half;

    v8f acc0 = {}, acc1 = {}, acc2 = {}, acc3 = {};
#pragma unroll 4
    for (int kk = 0; kk < Csz; kk += 4) {
        v2f a  = *(const v2f*)(arow + kk);
        v2f w0 = *(const v2f*)(b0 + kk);
        v2f w1 = *(const v2f*)(b1 + kk);
        v2f w2 = *(const v2f*)(b2 + kk);
        v2f w3 = *(const v2f*)(b3 + kk);
        acc0 = wmma_f32(a, w0, acc0);
        acc1 = wmma_f32(a, w1, acc1);
        acc2 = wmma_f32(a, w2, acc2);
        acc3 = wmma_f32(a, w3, acc3);
    }

    const size_t BTH = (size_t)Bsz * Tsz * Hsz;
    if (which < 2) {
        // D layout: lane col N = l16 (+16*nt); VGPR r -> row r + 8*half
        float* op = qkv + (size_t)which * BTH + (mrow + 8 * half) * (size_t)Hsz + l16;
#pragma unroll
        for (int r = 0; r < 8; ++r) {
            op[(size_t)r * Hsz +  0] = acc0[r];
            op[(size_t)r * Hsz + 16] = acc1[r];
            op[(size_t)r * Hsz + 32] = acc2[r];
            op[(size_t)r * Hsz + 48] = acc3[r];
        }
    } else {
        // V transposed: Vt[b][h][t]; consecutive r -> consecutive t.
        const long bb   = mrow / Tsz;
        const long tloc = mrow - bb * Tsz + 8 * half;
        float* vt = qkv + 2 * BTH;
        const v8f* accs[4] = { &acc0, &acc1, &acc2, &acc3 };
#pragma unroll
        for (int nt = 0; nt < 4; ++nt) {
            const int col = nt * 16 + l16;
            float* vp = vt + ((size_t)bb * Hsz + col) * Tsz + tloc;
            const v8f& a = *accs[nt];
            float4 lo = make_float4(a[0], a[1], a[2], a[3]);
            float4 hi = make_float4(a[4], a[5], a[6], a[7]);
            *(float4*)(vp + 0) = lo;     // 32B-aligned
            *(float4*)(vp + 4) = hi;
        }
    }
}

// ---------------------------------------------------------------------------
// Stage 2: flash attention, fp32 WMMA throughout.
// One wave (single-wave workgroup) per 16-row query tile; online softmax.
// grid = (T/16, B), block = 32.  Longest causal loops scheduled first.
// ---------------------------------------------------------------------------
__global__ __launch_bounds__(32) void attn_kernel(
    const float* __restrict__ qkv,
    float* __restrict__ out)    // [B, T, H]
{
    __shared__ float Plds[16 * 16];          // P transpose staging (1 KB)

    const int lane = threadIdx.x;
    const int half = lane >> 4;
    const int l16  = lane & 15;
    const int b    = blockIdx.y;
    const int qi   = gridDim.x - 1 - blockIdx.x;   // long loops first

    const size_t BTH = (size_t)Bsz * Tsz * Hsz;
    const float* Q  = qkv +            (size_t)b * Tsz * Hsz;   // [T][H]
    const float* K  = qkv + BTH     +  (size_t)b * Tsz * Hsz;   // [T][H]
    const float* Vt = qkv + 2 * BTH +  (size_t)b * Hsz * Tsz;   // [H][T]

    const float scale = 0.125f;              // 1/sqrt(64)
    const int qrow = qi * 16 + l16;

    // Preload & pre-scale the full 16x64 Q tile as 16 A-fragments (32 VGPRs).
    v2f qa[16];
#pragma unroll
    for (int c = 0; c < 16; ++c) {
        v2f t = *(const v2f*)(Q + (size_t)qrow * Hsz + 4 * c + 2 * half);
        qa[c][0] = t[0] * scale;
        qa[c][1] = t[1] * scale;
    }

    // Per-h-tile V row pointers (B-frag base: contiguous in t).
    const float* Vr0 = Vt + (size_t)( 0 + l16) * Tsz + 2 * half;
    const float* Vr1 = Vt + (size_t)(16 + l16) * Tsz + 2 * half;
    const float* Vr2 = Vt + (size_t)(32 + l16) * Tsz + 2 * half;
    const float* Vr3 = Vt + (size_t)(48 + l16) * Tsz + 2 * half;

    float mrow[8], lrow[8];                  // per-row running max / sum
#pragma unroll
    for (int r = 0; r < 8; ++r) { mrow[r] = -INFINITY; lrow[r] = 0.0f; }
    v8f o0 = {}, o1 = {}, o2 = {}, o3 = {};  // O accumulators: 16 x 64

    for (int kt = 0; kt <= qi; ++kt) {
        // ---- S = (Q*scale) @ K^T : 16 WMMAs over H=64 -------------------
        const float* Kt = K + (size_t)(kt * 16 + l16) * Hsz + 2 * half;
        v8f s = {};
#pragma unroll
        for (int c = 0; c < 16; ++c) {
            v2f kb = *(const v2f*)(Kt + 4 * c);   // B frag: col=key l16, K=h
            s = wmma_f32(qa[c], kb, s);
        }

        // ---- causal mask on diagonal tile ------------------------------
        if (kt == qi) {
#pragma unroll
            for (int r = 0; r < 8; ++r)
                if (l16 > r + 8 * half) s[r] = -INFINITY;
        }

        // ---- online softmax (row stats via 16-lane xor reductions) -----
        float alpha[8], p[8];
#pragma unroll
        for (int r = 0; r < 8; ++r) {
            float mx = s[r];
            mx = fmaxf(mx, __shfl_xor(mx, 1));
            mx = fmaxf(mx, __shfl_xor(mx, 2));
            mx = fmaxf(mx, __shfl_xor(mx, 4));
            mx = fmaxf(mx, __shfl_xor(mx, 8));
            const float mn = fmaxf(mrow[r], mx);
            alpha[r] = expf(mrow[r] - mn);
            mrow[r]  = mn;
            float pv = expf(s[r] - mn);
            p[r] = pv;
            float sm = pv;
            sm += __shfl_xor(sm, 1);
            sm += __shfl_xor(sm, 2);
            sm += __shfl_xor(sm, 4);
            sm += __shfl_xor(sm, 8);
            lrow[r] = lrow[r] * alpha[r] + sm;
        }

        // ---- transpose P through LDS into A-fragment layout ------------
#pragma unroll
        for (int r = 0; r < 8; ++r)
            Plds[(r + 8 * half) * 16 + l16] = p[r];
        __syncthreads();   // single-wave WG: compiler fence + in-order LDS

        // rescale running O by alpha (per row)
#pragma unroll
        for (int r = 0; r < 8; ++r) {
            o0[r] *= alpha[r]; o1[r] *= alpha[r];
            o2[r] *= alpha[r]; o3[r] *= alpha[r];
        }

        v2f pa[4];
#pragma unroll
        for (int c2 = 0; c2 < 4; ++c2)
            pa[c2] = *(const v2f*)(&Plds[l16 * 16 + 4 * c2 + 2 * half]);

        // ---- O += P @ V : contiguous b64 B-frag loads from Vt ----------
        const int kb0 = kt * 16;
#pragma unroll
        for (int c2 = 0; c2 < 4; ++c2) {
            const int ko = kb0 + 4 * c2;
            v2f vb;
            vb = *(const v2f*)(Vr0 + ko); o0 = wmma_f32(pa[c2], vb, o0);
            vb = *(const v2f*)(Vr1 + ko); o1 = wmma_f32(pa[c2], vb, o1);
            vb = *(const v2f*)(Vr2 + ko); o2 = wmma_f32(pa[c2], vb, o2);
            vb = *(const v2f*)(Vr3 + ko); o3 = wmma_f32(pa[c2], vb, o3);
        }
        __syncthreads();   // protect Plds WAR for next iteration
    }

    // ---- finalize: O / l, store [B,T,H] --------------------------------
    float* Ob = out + ((size_t)b * Tsz + qi * 16 + 8 * half) * Hsz + l16;
#pragma unroll
    for (int r = 0; r < 8; ++r) {
        const float inv = 1.0f / lrow[r];
        Ob[(size_t)r * Hsz +  0] = o0[r] * inv;
        Ob[(size_t)r * Hsz + 16] = o1[r] * inv;
        Ob[(size_t)r * Hsz + 32] = o2[r] * inv;
        Ob[(size_t)r * Hsz + 48] = o3[r] * inv;
    }
}

extern "C" void kernel_launch(void* const* d_in, const int* in_sizes, int n_in,
                              void* d_out, int out_size, void* d_ws, size_t ws_size,
                              hipStream_t stream) {
    const float* x  = (const float*)d_in[0];
    const float* Wq = (const float*)d_in[1];
    const float* Wk = (const float*)d_in[2];
    const float* Wv = (const float*)d_in[3];

    const size_t BTH = (size_t)Bsz * Tsz * Hsz;
    float* qkv = (float*)d_ws;          // 3 * B*T*H floats = 12 MB
    float* Wt  = qkv + 3 * BTH;         // 3 * H*C floats   = 768 KB

    dim3 g0(Hsz, 3), blk0(256);
    wt_kernel<<<g0, blk0, 0, stream>>>(Wq, Wk, Wv, Wt);

    dim3 g1(Bsz * Tsz / 64, 3), blk1(128);
    qkv_kernel<<<g1, blk1, 0, stream>>>(x, Wt, qkv);

    dim3 g2(Tsz / 16, Bsz), blk2(32);
    attn_kernel<<<g2, blk2, 0, stream>>>(qkv, (float*)d_out);
}